// WarpController_12034498363858
// MI455X (gfx1250) — compile-verified
//
#include <hip/hip_runtime.h>
#include <hip/hip_bf16.h>

// ---------------- problem constants ----------------
#define BATCH   8
#define KTOT    196608      // 3*256*256
#define NOUT    128
#define NLD     136         // 68 landmarks * 2
#define NL      68
#define NA      71          // 68 + 3 (affine)
#define NC      73          // 71 + 2 rhs cols
#define HW      65536       // 256*256
#define CHANS   64

// GEMM1 split-K config
#define KBLOCKS 128
#define KPB     (KTOT / KBLOCKS)   // 1536
#define CK      64                 // LDS-staged sub-chunk
#define SUBS    (KPB / CK)         // 24
#define LDP     66                 // padded LDS row stride (floats)

typedef __attribute__((ext_vector_type(2))) float v2f;
typedef __attribute__((ext_vector_type(8))) float v8f;

// =====================================================================
// Kernel 1: h_partial = x(8xK) @ W1^T(Kx128), split-K, fp32 WMMA 16x16x4
// =====================================================================
__global__ __launch_bounds__(256) void gemm1_wmma(
    const float* __restrict__ x, const float* __restrict__ W1,
    float* __restrict__ part) {
  __shared__ float xs[16][LDP];     // A tile: rows 8..15 stay zero (M pad)
  __shared__ float wsm[128][LDP];   // B tile: all 8 N-tiles (128 rows of W1)

  const int tid  = threadIdx.x;
  const int lane = tid & 31;
  const int wv   = tid >> 5;        // wave id = N-tile id (0..7)
  const int half = lane >> 4;       // 0: K=(0,1)  1: K=(2,3)
  const int l16  = lane & 15;
  const int kb   = blockIdx.x;
  const int kbase0 = kb * KPB;

  // zero the padded A rows (M = 8..15) once
  for (int i = tid; i < 8 * LDP; i += 256) xs[8 + i / LDP][i % LDP] = 0.0f;

  v8f c = {};
  for (int s = 0; s < SUBS; ++s) {
    const int kbase = kbase0 + s * CK;
    __syncthreads();   // protect previous iteration's LDS reads
    // stage x chunk: 8 rows x 64 cols (coalesced)
    for (int i = tid; i < 8 * CK; i += 256) {
      int r = i >> 6, cc = i & (CK - 1);
      xs[r][cc] = x[r * KTOT + kbase + cc];
    }
    // stage W1 chunk: 128 rows x 64 cols (coalesced)
    for (int i = tid; i < 128 * CK; i += 256) {
      int r = i >> 6, cc = i & (CK - 1);
      wsm[r][cc] = W1[r * KTOT + kbase + cc];
    }
    // prefetch next chunk of W1 into cache (gfx1250 global_prefetch_b8)
    if (s + 1 < SUBS) {
      int r = tid >> 1;             // 128 rows, 2 threads/row
      __builtin_prefetch(&W1[r * KTOT + kbase + CK + (tid & 1) * 32], 0, 0);
    }
    __syncthreads();
    // 16 WMMA issues cover K=64:  C(16x16) += A(16x4) * B(4x16)
    #pragma unroll
    for (int it = 0; it < CK / 4; ++it) {
      const int k0 = it * 4 + half * 2;
      v2f a = *(const v2f*)&xs[l16][k0];                // A[m=l16][k0,k0+1]
      v2f b = *(const v2f*)&wsm[wv * 16 + l16][k0];     // B[k][n]=W1[n][k]
      c = __builtin_amdgcn_wmma_f32_16x16x4_f32(
          false, a, false, b, (short)0, c, false, false);
    }
  }
  // C layout: VGPR r -> (M=r, N=lane) for lanes 0..15 ; M=8..15 are padding
  if (lane < 16) {
    float* dst = part + kb * 1024 + wv * 16 + lane;
    #pragma unroll
    for (int r = 0; r < 8; ++r) dst[r * 128] = c[r];
  }
}

// =====================================================================
// Kernel 2: reduce split-K partials -> h ; landmark heads ; norm
// =====================================================================
__global__ __launch_bounds__(256) void landmarks_kernel(
    const float* __restrict__ part, const float* __restrict__ b1,
    const float* __restrict__ W2, const float* __restrict__ b2,
    const float* __restrict__ W3, const float* __restrict__ b3,
    const float* __restrict__ lmean, const float* __restrict__ scales,
    float* __restrict__ out_pred, float* __restrict__ out_norm,
    float* __restrict__ dstp, float* __restrict__ flw) {
  __shared__ float h[1024];
  __shared__ float sq[8];
  const int tid = threadIdx.x;

  for (int i = tid; i < 1024; i += 256) {         // i = m*128 + n
    float s = b1[i & 127];
    for (int kb = 0; kb < KBLOCKS; ++kb) s += part[kb * 1024 + i];
    h[i] = s;
  }
  __syncthreads();

  for (int idx = tid; idx < BATCH * NLD; idx += 256) {
    const int m = idx / NLD, j = idx % NLD;
    const float* hm = &h[m * 128];
    const float* w2 = &W2[j * 128];
    const float* w3 = &W3[j * 128];
    float p = b2[j], d = b3[j];
    for (int cdx = 0; cdx < 128; ++cdx) {
      p += hm[cdx] * w2[cdx];
      d += hm[cdx] * w3[cdx];
    }
    p += lmean[j];
    d *= scales[m];
    out_pred[idx] = p;          // landmarks_pred
    dstp[idx]     = p + d;      // dst points (spline train points)
    flw[idx]      = d;          // flows (spline train values)
  }
  __syncthreads();

  if (tid < BATCH) {            // deterministic norm reduction
    float s = 0.f;
    for (int j = 0; j < NLD; ++j) { float dd = flw[tid * NLD + j]; s += dd * dd; }
    sq[tid] = sqrtf(s);
  }
  __syncthreads();
  if (tid == 0) {
    float s = 0.f;
    for (int m = 0; m < BATCH; ++m) s += sq[m];
    out_norm[0] = s * 0.125f;
  }
}

// =====================================================================
// Kernel 3: per-batch 71x71 TPS system, Gauss-Jordan + partial pivoting
// =====================================================================
__global__ __launch_bounds__(128) void spline_solve(
    const float* __restrict__ dstp, const float* __restrict__ flw,
    float* __restrict__ sol) {
  __shared__ float M[NA][NC + 1];   // augmented, padded stride
  __shared__ float pts[NL][2];
  __shared__ int prow_s;
  const int b = blockIdx.x, tid = threadIdx.x;

  for (int i = tid; i < NL * 2; i += 128) pts[i >> 1][i & 1] = dstp[b * NLD + i];
  __syncthreads();

  for (int idx = tid; idx < NA * NC; idx += 128) {
    const int i = idx / NC, j = idx % NC;
    float v = 0.f;
    if (i < NL && j < NL) {                       // A = phi(d^2) + reg*I
      float dy = pts[i][0] - pts[j][0];
      float dx = pts[i][1] - pts[j][1];
      float r = dy * dy + dx * dx;
      v = 0.5f * r * __logf(fmaxf(r, 1e-10f));
      if (i == j) v += 1e-6f;
    } else if (i < NL && j < NA) {                // B block
      v = (j == NL + 2) ? 1.0f : pts[i][j - NL];
    } else if (i < NL) {                          // rhs = flows
      v = flw[b * NLD + i * 2 + (j - NA)];
    } else if (j < NL) {                          // B^T block
      v = (i == NL + 2) ? 1.0f : pts[j][i - NL];
    }                                             // else 0
    M[i][j] = v;
  }
  __syncthreads();

  for (int k = 0; k < NA; ++k) {
    if (tid == 0) {                               // partial pivot search
      int pr = k; float best = fabsf(M[k][k]);
      for (int i = k + 1; i < NA; ++i) {
        float av = fabsf(M[i][k]);
        if (av > best) { best = av; pr = i; }
      }
      prow_s = pr;
    }
    __syncthreads();
    const int pr = prow_s;
    for (int j = tid; j < NC; j += 128) {         // row swap (self-swap ok)
      float t = M[k][j]; M[k][j] = M[pr][j]; M[pr][j] = t;
    }
    __syncthreads();
    const float inv = 1.0f / M[k][k];
    __syncthreads();
    for (int j = tid; j < NC; j += 128) M[k][j] *= inv;
    __syncthreads();
    for (int i = tid; i < NA; i += 128) {
      if (i != k) {
        const float f = M[i][k];
        for (int j = k; j < NC; ++j) M[i][j] -= f * M[k][j];
      }
    }
    __syncthreads();
  }
  for (int i = tid; i < NA * 2; i += 128)         // sol: w rows 0..67, v 68..70
    sol[b * (NA * 2) + i] = M[i >> 1][NA + (i & 1)];
}

// =====================================================================
// Kernel 4: fused per-pixel spline flow + 64-channel bilinear warp
// =====================================================================
__global__ __launch_bounds__(256) void flow_warp(
    const float* __restrict__ sol, const float* __restrict__ dstp,
    const float* __restrict__ img, float* __restrict__ out) {
  __shared__ float wL[NL][2];
  __shared__ float vL[3][2];
  __shared__ float ptL[NL][2];
  const int b = blockIdx.y;
  const int tid = threadIdx.x;

  if (tid < NA * 2) {
    float v = sol[b * (NA * 2) + tid];
    int i = tid >> 1, d = tid & 1;
    if (i < NL) wL[i][d] = v; else vL[i - NL][d] = v;
  }
  if (tid < NL * 2) ptL[tid >> 1][tid & 1] = dstp[b * NLD + tid];
  __syncthreads();

  const int p = blockIdx.x * 256 + tid;
  const float fy = (float)(p >> 8);
  const float fx = (float)(p & 255);
  float f0 = vL[0][0] * fy + vL[1][0] * fx + vL[2][0];
  float f1 = vL[0][1] * fy + vL[1][1] * fx + vL[2][1];
  #pragma unroll 4
  for (int i = 0; i < NL; ++i) {
    float dy = fy - ptL[i][0];
    float dx = fx - ptL[i][1];
    float r = dy * dy + dx * dx;
    float ph = 0.5f * r * __logf(fmaxf(r, 1e-10f));
    f0 += ph * wL[i][0];
    f1 += ph * wL[i][1];
  }
  const float qy = fminf(fmaxf(fy - f0, 0.f), 255.f);
  const float qx = fminf(fmaxf(fx - f1, 0.f), 255.f);
  int y0 = (int)floorf(qy); y0 = y0 > 254 ? 254 : y0;
  int x0 = (int)floorf(qx); x0 = x0 > 254 ? 254 : x0;
  const float wy = qy - (float)y0, wx = qx - (float)x0;
  const float w00 = (1.f - wx) * (1.f - wy), w01 = wx * (1.f - wy);
  const float w10 = (1.f - wx) * wy,         w11 = wx * wy;
  const int i00 = y0 * 256 + x0;
  const float* base = img + b * (CHANS * HW);
  float* ob = out + b * (CHANS * HW) + p;
  for (int c = 0; c < CHANS; ++c) {
    const float* pc = base + c * HW;
    float v00 = pc[i00], v01 = pc[i00 + 1];
    float v10 = pc[i00 + 256], v11 = pc[i00 + 257];
    ob[c * HW] = v00 * w00 + v01 * w01 + v10 * w10 + v11 * w11;
  }
}

// =====================================================================
extern "C" void kernel_launch(void* const* d_in, const int* in_sizes, int n_in,
                              void* d_out, int out_size, void* d_ws, size_t ws_size,
                              hipStream_t stream) {
  (void)in_sizes; (void)n_in; (void)out_size; (void)ws_size;
  const float* x      = (const float*)d_in[0];
  const float* img    = (const float*)d_in[1];
  const float* scales = (const float*)d_in[2];
  const float* lmean  = (const float*)d_in[3];
  const float* W1     = (const float*)d_in[4];
  const float* b1     = (const float*)d_in[5];
  const float* W2     = (const float*)d_in[6];
  const float* b2     = (const float*)d_in[7];
  const float* W3     = (const float*)d_in[8];
  const float* b3     = (const float*)d_in[9];
  float* out = (float*)d_out;
  float* ws  = (float*)d_ws;

  // workspace layout (floats): part | dstp | flw | sol  (~0.52 MB total)
  float* part = ws;                   // 128 * 1024
  float* dstp = ws + 131072;          // 8 * 136
  float* flw  = ws + 132160;          // 8 * 136
  float* sol  = ws + 133248;          // 8 * 142

  float* out_warp = out;                       // 8*64*256*256
  float* out_pred = out + BATCH * CHANS * HW;  // 8*136
  float* out_norm = out_pred + BATCH * NLD;    // 1

  hipLaunchKernelGGL(gemm1_wmma, dim3(KBLOCKS), dim3(256), 0, stream,
                     x, W1, part);
  hipLaunchKernelGGL(landmarks_kernel, dim3(1), dim3(256), 0, stream,
                     part, b1, W2, b2, W3, b3, lmean, scales,
                     out_pred, out_norm, dstp, flw);
  hipLaunchKernelGGL(spline_solve, dim3(BATCH), dim3(128), 0, stream,
                     dstp, flw, sol);
  hipLaunchKernelGGL(flow_warp, dim3(HW / 256, BATCH), dim3(256), 0, stream,
                     sol, dstp, img, out_warp);
}